// TGAN_84885733638248
// MI455X (gfx1250) — compile-verified
//
#include <hip/hip_runtime.h>

typedef __attribute__((ext_vector_type(2))) float v2f;
typedef __attribute__((ext_vector_type(8))) float v8f;

#define FDIM 128
#define KCAT 512   // (K+1)*F = 4*128

// ---------------- degree / norm ----------------
__global__ void tag_deg_kernel(const long long* __restrict__ dst, float* __restrict__ deg, int E) {
    int i = blockIdx.x * blockDim.x + threadIdx.x;
    if (i < E) atomicAdd(&deg[(int)dst[i]], 1.0f);
}

__global__ void tag_dinv_kernel(float* __restrict__ d, int N) {
    int i = blockIdx.x * blockDim.x + threadIdx.x;
    if (i < N) {
        float v = d[i];
        d[i] = (v > 0.0f) ? rsqrtf(fmaxf(v, 1.0f)) : 0.0f;
    }
}

__global__ void tag_norm_kernel(const long long* __restrict__ src, const long long* __restrict__ dst,
                                const float* __restrict__ dinv, float* __restrict__ norm, int E) {
    int i = blockIdx.x * blockDim.x + threadIdx.x;
    if (i < E) norm[i] = dinv[(int)src[i]] * dinv[(int)dst[i]];
}

// ---------------- SpMM: hout[dst] += norm * hin[src], width 128 ----------------
// one wave per edge; each lane handles 4 consecutive floats (float4)
__global__ void tag_spmm_kernel(const long long* __restrict__ src, const long long* __restrict__ dst,
                                const float* __restrict__ norm,
                                const float* __restrict__ hin, float* __restrict__ hout, int E) {
    long long tid = (long long)blockIdx.x * blockDim.x + threadIdx.x;
    long long e = tid >> 5;
    if (e >= E) return;
    int c = ((int)tid & 31) * 4;
    int s = (int)src[e];
    int d = (int)dst[e];
    float w = norm[e];
    const float4 v = *(const float4*)(hin + (long long)s * FDIM + c);
    float* o = hout + (long long)d * FDIM + c;
    atomicAdd(o + 0, w * v.x);
    atomicAdd(o + 1, w * v.y);
    atomicAdd(o + 2, w * v.z);
    atomicAdd(o + 3, w * v.w);
}

// ---------------- WMMA GEMM: out[16n x 128] = cat_hops[16n x 512] @ W[512 x 128] + b ----------------
// hops = 4 contiguous [N x 128] blocks; logical column k -> hop k>>7, col k&127.
// One wave computes a 16-node x 128-col strip with V_WMMA_F32_16X16X4_F32.
template <bool RELU>
__global__ __launch_bounds__(256) void tag_gemm_kernel(const float* __restrict__ hops,
                                                       const float* __restrict__ W,
                                                       const float* __restrict__ bias,
                                                       float* __restrict__ out,
                                                       int N) {
    const int lane = threadIdx.x & 31;
    const int wave = threadIdx.x >> 5;
    const int tile = blockIdx.x * 8 + wave;
    const int ntiles = N >> 4;
    if (tile >= ntiles) return;

    const int hi   = lane >> 4;     // half-wave select
    const int ncol = lane & 15;     // column within 16-wide tile
    const int m    = (tile << 4) + ncol;  // A-matrix row for this lane (row m = lane%16)
    const long long NF = (long long)N * FDIM;

    v8f acc[8];
    for (int j = 0; j < 8; ++j) {
        float bv = bias[j * 16 + ncol];
        for (int v = 0; v < 8; ++v) acc[j][v] = bv;
    }

#pragma unroll 1
    for (int k0 = 0; k0 < KCAT; k0 += 4) {
        const int ka = k0 + 2 * hi;   // even; VGPR0 holds K=ka, VGPR1 holds K=ka+1 (ISA 16x4 A layout)
        // ka and ka+1 are in the same 128-wide hop block (ka even) -> single aligned 8B load
        const v2f a = *(const v2f*)(hops + (long long)(ka >> 7) * NF
                                         + (long long)m * FDIM + (ka & 127));
        v2f bf[8];
        for (int j = 0; j < 8; ++j) {
            bf[j].x = W[(long long)(ka)     * FDIM + j * 16 + ncol];
            bf[j].y = W[(long long)(ka + 1) * FDIM + j * 16 + ncol];
        }
        for (int j = 0; j < 8; ++j) {
            acc[j] = __builtin_amdgcn_wmma_f32_16x16x4_f32(
                false, a, false, bf[j], (short)0, acc[j], false, false);
        }
    }

    const int mrow0 = tile << 4;
    for (int j = 0; j < 8; ++j) {
        for (int v = 0; v < 8; ++v) {
            float r = acc[j][v];
            if (RELU) r = fmaxf(r, 0.0f);
            // D layout: row = v + 8*hi, col = j*16 + ncol
            out[(long long)(mrow0 + v + 8 * hi) * FDIM + j * 16 + ncol] = r;
        }
    }
}

// scalar tail for N not multiple of 16 (unused when N%16==0, kept for safety)
__global__ void tag_gemm_tail_kernel(const float* __restrict__ hops, const float* __restrict__ W,
                                     const float* __restrict__ bias, float* __restrict__ out,
                                     int N, int base, int relu) {
    int t = blockIdx.x * blockDim.x + threadIdx.x;
    int rem = N - base;
    if (t >= rem * FDIM) return;
    int node = base + t / FDIM;
    int col  = t % FDIM;
    long long NF = (long long)N * FDIM;
    float s = bias[col];
    for (int k = 0; k < KCAT; ++k)
        s += hops[(long long)(k >> 7) * NF + (long long)node * FDIM + (k & 127)] * W[(long long)k * FDIM + col];
    if (relu) s = fmaxf(s, 0.0f);
    out[(long long)node * FDIM + col] = s;
}

// ---------------- final GEMV: out[n] = cat_hops[n,:] . W2 + b2 ----------------
__global__ void tag_dot_kernel(const float* __restrict__ hops, const float* __restrict__ W2,
                               const float* __restrict__ b2, float* __restrict__ out, int N) {
    long long tid = (long long)blockIdx.x * blockDim.x + threadIdx.x;
    int node = (int)(tid >> 5);
    if (node >= N) return;
    int lane = (int)tid & 31;
    long long NF = (long long)N * FDIM;
    float s = 0.0f;
    for (int i = 0; i < KCAT / 32; ++i) {
        int k = i * 32 + lane;
        s += hops[(long long)(k >> 7) * NF + (long long)node * FDIM + (k & 127)] * W2[k];
    }
    for (int off = 16; off; off >>= 1) s += __shfl_xor(s, off, 32);
    if (lane == 0) out[node] = s + b2[0];
}

extern "C" void kernel_launch(void* const* d_in, const int* in_sizes, int n_in,
                              void* d_out, int out_size, void* d_ws, size_t ws_size,
                              hipStream_t stream) {
    const float*     x   = (const float*)d_in[0];
    const long long* ei  = (const long long*)d_in[1];
    const float*     W1  = (const float*)d_in[2];
    const float*     b1  = (const float*)d_in[3];
    const float*     W2  = (const float*)d_in[4];
    const float*     b2  = (const float*)d_in[5];

    const int N = in_sizes[0] / FDIM;
    const int E = in_sizes[1] / 2;
    const long long* src = ei;
    const long long* dst = ei + E;
    const long long NF = (long long)N * FDIM;

    float* ws   = (float*)d_ws;
    float* dinv = ws;                  // N floats (deg, then dinv in place)
    float* norm = dinv + N;            // E floats
    float* hopA = norm + E;            // 4*N*F floats (layer-1 hops)
    float* hopB = hopA + 4 * NF;       // 4*N*F floats (layer-2 hops; slot0 = relu(GEMM1))

    // ---- gcn norm ----
    hipMemsetAsync(dinv, 0, (size_t)N * sizeof(float), stream);
    tag_deg_kernel <<<(E + 255) / 256, 256, 0, stream>>>(dst, dinv, E);
    tag_dinv_kernel<<<(N + 255) / 256, 256, 0, stream>>>(dinv, N);
    tag_norm_kernel<<<(E + 255) / 256, 256, 0, stream>>>(src, dst, dinv, norm, E);

    // ---- layer-1 hop chain ----
    hipMemcpyAsync(hopA, x, (size_t)NF * sizeof(float), hipMemcpyDeviceToDevice, stream);
    const long long spmm_threads = (long long)E * 32;
    const int spmm_blocks = (int)((spmm_threads + 255) / 256);
    for (int k = 1; k < 4; ++k) {
        float* hin  = hopA + (long long)(k - 1) * NF;
        float* hout = hopA + (long long)k * NF;
        hipMemsetAsync(hout, 0, (size_t)NF * sizeof(float), stream);
        tag_spmm_kernel<<<spmm_blocks, 256, 0, stream>>>(src, dst, norm, hin, hout, E);
    }

    // ---- layer-1 dense: hopB[0] = relu(cat(hopA) @ W1 + b1) ----
    const int ntiles = N >> 4;
    if (ntiles > 0)
        tag_gemm_kernel<true><<<(ntiles + 7) / 8, 256, 0, stream>>>(hopA, W1, b1, hopB, N);
    if (N & 15) {
        int base = ntiles << 4;
        int rem  = N - base;
        tag_gemm_tail_kernel<<<(rem * FDIM + 255) / 256, 256, 0, stream>>>(hopA, W1, b1, hopB, N, base, 1);
    }

    // ---- layer-2 hop chain ----
    for (int k = 1; k < 4; ++k) {
        float* hin  = hopB + (long long)(k - 1) * NF;
        float* hout = hopB + (long long)k * NF;
        hipMemsetAsync(hout, 0, (size_t)NF * sizeof(float), stream);
        tag_spmm_kernel<<<spmm_blocks, 256, 0, stream>>>(src, dst, norm, hin, hout, E);
    }

    // ---- layer-2 GEMV -> d_out ----
    const long long dot_threads = (long long)N * 32;
    tag_dot_kernel<<<(int)((dot_threads + 255) / 256), 256, 0, stream>>>(hopB, W2, b2, (float*)d_out, N);
}